// LohaloKernelScalingSampler_57208964383227
// MI455X (gfx1250) — compile-verified
//
#include <hip/hip_runtime.h>
#include <math.h>

// ---------------------------------------------------------------------------
// LoHalo scaling sampler for MI455X (gfx1250, wave32).
// image: (1,3,1024,1024) f32   grid: (1,768,768,2) f32   out: (1,3,768,768) f32
//
// Mitchell stage runs through V_WMMA_F32_16X16X4_F32 (fp32-exact):
//   A[i][k] = sig(pixel)*mitchell_weight for pixel i, tap chunk k ; B = ones
//   => every column of D is the 36-tap weighted sum (chained over 9 chunks).
// EWA stage is skipped per-wave via __any(need_ewa) (uniform branch).
// All divisions / logs / tanh use raw v_rcp/v_log/v_exp/v_rsq hardware ops.
// Piecewise sigmoids use 0/1-mask arithmetic blending (like the reference) so
// clang cannot turn them into per-value EXEC-masked branches.
// ---------------------------------------------------------------------------

typedef float v2f __attribute__((ext_vector_type(2)));
typedef float v8f __attribute__((ext_vector_type(8)));

#define H_IN   1024
#define W_IN   1024
#define H_OUT  768
#define W_OUT  768
#define NCH    3
#define WIN    6          // 2*WIN_R
#define NTAP   36
#define CONTRAST 3.38589f
#define SQRT2D 1.4142135623730951
#define LN2F   0.69314718055994531f
#define LOG2EF 1.4426950408889634f

__device__ __forceinline__ int clampi(int v, int lo, int hi) {
    return v < lo ? lo : (v > hi ? hi : v);
}

__device__ __forceinline__ float mitchell(float x) {
    float ax  = fabsf(x);
    float ax2 = ax * ax;
    float ax3 = ax2 * ax;
    float v1  = (7.0f / 6.0f) * ax3 - 2.0f * ax2 + (8.0f / 9.0f);
    float v2  = (-7.0f / 18.0f) * ax3 + 2.0f * ax2 - (10.0f / 3.0f) * ax + (16.0f / 9.0f);
    return (ax < 1.0f) ? v1 : ((ax < 2.0f) ? v2 : 0.0f);
}

__device__ __forceinline__ float robidoux(float r2) {
    const float a3   = -3.0f;
    const float a2   = (float)((45739.0 + 7164.0 * SQRT2D) / 10319.0);
    const float a0   = (float)((-8926.0 - 14328.0 * SQRT2D) / 10319.0);
    const float m_in = (float)((-103.0 - 36.0 * SQRT2D) / (7.0 + 72.0 * SQRT2D));
    float r     = __builtin_amdgcn_sqrtf(r2 + 1e-8f);
    float w_in  = r2 * (a3 * r + a2) + a0;
    float t     = r - 2.0f;                 // (r + m_out_root), m_out_root = -2
    float w_out = (r + m_in) * t * t;
    return (r2 < 1.0f) ? w_in : ((r2 < 4.0f) ? w_out : 0.0f);
}

// (2/c)*atanh(z)+0.5 with z clamped to +-0.999999:
// ratio = (1+z)/(1-z) in [5e-7, 2e6] -> raw v_log_f32 (log2) is safe.
// Branchless mask blend (m_lo/m_hi/m_mid as in the reference) so the log path
// is never placed behind a divergent EXEC branch.
__device__ __forceinline__ float inv_sigmoid(float q, float sig1, float inv) {
    float z     = fminf(fmaxf(2.0f * sig1 * q - sig1, -0.999999f), 0.999999f);
    float ratio = (1.0f + z) * __builtin_amdgcn_rcpf(1.0f - z);
    float mid   = (LN2F / CONTRAST) * __builtin_amdgcn_logf(ratio) + 0.5f;
    float m_lo  = (q <= 0.0f) ? 1.0f : 0.0f;
    float m_hi  = (q >= 1.0f) ? 1.0f : 0.0f;
    float m_mid = 1.0f - m_lo - m_hi;
    // q*inv*m_lo + (q*inv + 1 - inv)*m_hi + mid*m_mid
    return q * inv * (m_lo + m_hi) + (1.0f - inv) * m_hi + mid * m_mid;
}

// tanh(u) = 1 - 2/(exp2(2*log2e*u)+1); u in ~[-3,3] here -> exp2 arg ~[-9,9].
__device__ __forceinline__ float ext_sigmoid(float q, float rsig1h, float slope) {
    float u     = 0.5f * CONTRAST * q - 0.25f * CONTRAST;
    float e     = __builtin_amdgcn_exp2f(2.0f * LOG2EF * u);
    float th    = 1.0f - 2.0f * __builtin_amdgcn_rcpf(e + 1.0f);
    float mid   = rsig1h * th + 0.5f;
    float m_lo  = (q <= 0.0f) ? 1.0f : 0.0f;
    float m_hi  = (q >= 1.0f) ? 1.0f : 0.0f;
    float m_mid = 1.0f - m_lo - m_hi;
    return slope * q * (m_lo + m_hi) + (1.0f - slope) * m_hi + mid * m_mid;
}

// Pick the diagonal element D(i,i) for pixel i = (lane&15) of this group.
// C/D f32 16x16 layout: VGPR r holds M=r (lanes 0-15) / M=8+r (lanes 16-31), N=lane%16.
// Diagonal lives at: i<8 -> (vgpr i, lane i); i>=8 -> (vgpr i-8, lane i+16).
__device__ __forceinline__ float diag_extract(v8f acc, int lane) {
    int idx = lane & 7;
    float v = acc[0];
    v = (idx == 1) ? acc[1] : v;
    v = (idx == 2) ? acc[2] : v;
    v = (idx == 3) ? acc[3] : v;
    v = (idx == 4) ? acc[4] : v;
    v = (idx == 5) ? acc[5] : v;
    v = (idx == 6) ? acc[6] : v;
    v = (idx == 7) ? acc[7] : v;
    int p    = lane & 15;
    int srcx = (p < 8) ? p : (p + 16);
    return __shfl(v, srcx, 32);
}

__global__ __launch_bounds__(256)
void LohaloKernelScalingSampler_57208964383227_kernel(const float* __restrict__ img,
                                                      const float* __restrict__ g,
                                                      float* __restrict__ out) {
    const int tid  = threadIdx.x;
    const int x    = blockIdx.x * 256 + tid;
    const int y    = blockIdx.y;
    const int lane = tid & 31;

    // Sigmoid family constants (constant-folded by the compiler).
    const float sig1   = tanhf(0.25f * CONTRAST);
    const float rsig1  = __builtin_amdgcn_rcpf(sig1);
    const float slope  = (rsig1 - sig1) * 0.25f * CONTRAST;
    const float inv    = __builtin_amdgcn_rcpf(slope);
    const float rsig1h = 0.5f * rsig1;

    // ---- Jacobian from central differences with edge clamping --------------
    const float2* g2 = (const float2*)g;
    const int xm = clampi(x - 1, 0, W_OUT - 1), xp = clampi(x + 1, 0, W_OUT - 1);
    const int ym = clampi(y - 1, 0, H_OUT - 1), yp = clampi(y + 1, 0, H_OUT - 1);
    const float2 gc  = g2[y * W_OUT + x];
    const float2 gxl = g2[y * W_OUT + xm];
    const float2 gxr = g2[y * W_OUT + xp];
    const float2 gyd = g2[ym * W_OUT + x];
    const float2 gyu = g2[yp * W_OUT + x];
    const float J00 = (gxr.x - gxl.x) * 0.5f;
    const float J10 = (gxr.y - gxl.y) * 0.5f;
    const float J01 = (gyu.x - gyd.x) * 0.5f;
    const float J11 = (gyu.y - gyd.y) * 0.5f;

    const float det  = J00 * J11 - J01 * J10 + 1e-8f;
    const float rdet = __builtin_amdgcn_rcpf(det);
    const float a  =  J11 * rdet;
    const float b  = -J01 * rdet;
    const float cc = -J10 * rdet;
    const float d  =  J00 * rdet;
    const float n11 = a * a + b * b;
    const float n12 = a * cc + b * d;
    const float n22 = cc * cc + d * d;
    const float frob = n11 + n22;
    const float disc = frob * frob - 4.0f * rdet * rdet;
    const float sqrt_disc  = __builtin_amdgcn_sqrtf(fmaxf(disc, 0.0f));
    const float twice_s1s1 = frob + sqrt_disc;
    const float s1s1 = 0.5f * twice_s1s1;
    const float s2s2 = 0.5f * (frob - sqrt_disc);
    const float rmaj = __builtin_amdgcn_rsqf(fmaxf(s1s1, 1.0f));  // 1/major_mag
    const float rmin = __builtin_amdgcn_rsqf(fmaxf(s2s2, 1.0f));  // 1/minor_mag
    const float diff1 = s1s1 - n11;
    const float diff2 = s1s1 - n22;
    const bool  cond  = diff1 * diff1 >= diff2 * diff2;
    const float tu11 = cond ? n12 : diff2;
    const float tu21 = cond ? diff1 : n12;
    const float nrm2 = tu11 * tu11 + tu21 * tu21;
    const float rnrm = __builtin_amdgcn_rsqf(nrm2);
    const float u11  = (nrm2 > 0.0f) ? tu11 * rnrm : 1.0f;
    const float u21  = (nrm2 > 0.0f) ? tu21 * rnrm : 0.0f;
    const float cmaj_x = u11 * rmaj, cmaj_y = u21 * rmaj;
    const float cmin_x = -u21 * rmin, cmin_y = u11 * rmin;
    const float theta  = rmaj * rmin;
    const bool  need_ewa = twice_s1s1 > 2.0f;
    const bool  any_ewa  = __any((int)need_ewa) != 0;   // wave-uniform

    // ---- Window setup ------------------------------------------------------
    const int ix = (int)floorf(gc.x);
    const int iy = (int)floorf(gc.y);
    const float fx = gc.x - ((float)ix + 0.5f);
    const float fy = gc.y - ((float)iy + 0.5f);

    float relx[WIN], rely[WIN], mx[WIN], my[WIN];
    int   cols[WIN], rows[WIN];
#pragma unroll
    for (int k = 0; k < WIN; k++) {
        float o = (float)(k - 2);            // offsets -2..3
        relx[k] = fx - o;  rely[k] = fy - o;
        mx[k] = mitchell(relx[k]);
        my[k] = mitchell(rely[k]);
        cols[k] = clampi(ix + (k - 2), 0, W_IN - 1);
        rows[k] = clampi(iy + (k - 2), 0, H_IN - 1);
    }

    // ---- Mitchell stage via chained fp32 WMMA ------------------------------
    // Per group g (pixels of lanes g*16..g*16+15): A[i][k] = sig*wm for chunk
    // taps; B = ones => D row-sums accumulate the 36-tap weighted sum.
    const int rowM = lane & 15;
    const int kk   = (lane >> 4) & 1;        // 0: K=0,1 ; 1: K=2,3 (A layout)
    const v2f bones = {1.0f, 1.0f};

    v8f acc[2][NCH];
#pragma unroll
    for (int gi = 0; gi < 2; gi++)
#pragma unroll
        for (int ch = 0; ch < NCH; ch++)
            acc[gi][ch] = (v8f){0, 0, 0, 0, 0, 0, 0, 0};

    float sb[NCH][4];                        // 4-tap staging (registers)

#pragma unroll
    for (int t = 0; t < NTAP; t++) {
        const int j = t / WIN, k = t % WIN;
        const int base = rows[j] * W_IN + cols[k];
        const float p0 = img[base];
        const float p1 = img[base + H_IN * W_IN];
        const float p2 = img[base + 2 * H_IN * W_IN];
        const float wm = mx[k] * my[j];
        sb[0][t & 3] = inv_sigmoid(p0, sig1, inv) * wm;
        sb[1][t & 3] = inv_sigmoid(p1, sig1, inv) * wm;
        sb[2][t & 3] = inv_sigmoid(p2, sig1, inv) * wm;

        if ((t & 3) == 3) {                  // chunk of K=4 complete
#pragma unroll
            for (int gi = 0; gi < 2; gi++) {
                const int src = gi * 16 + rowM;
#pragma unroll
                for (int ch = 0; ch < NCH; ch++) {
                    // A layout 16x4 f32: lane m -> A[m][0..1], lane m+16 -> A[m][2..3]
                    float a0 = __shfl(sb[ch][0], src, 32);
                    float a1 = __shfl(sb[ch][1], src, 32);
                    float a2 = __shfl(sb[ch][2], src, 32);
                    float a3 = __shfl(sb[ch][3], src, 32);
                    v2f A;
                    A.x = kk ? a2 : a0;
                    A.y = kk ? a3 : a1;
                    acc[gi][ch] = __builtin_amdgcn_wmma_f32_16x16x4_f32(
                        false, A, false, bones, (short)0, acc[gi][ch], false, false);
                }
            }
        }
    }

    // Diagonal extraction + group select.
    float msum[NCH];
#pragma unroll
    for (int ch = 0; ch < NCH; ch++) {
        float m0 = diag_extract(acc[0][ch], lane);
        float m1 = diag_extract(acc[1][ch], lane);
        msum[ch] = (lane < 16) ? m0 : m1;
    }

    // ---- EWA stage (wave-uniform skip; ~always skipped for this grid) ------
    float ewa[NCH] = {0.0f, 0.0f, 0.0f};
    float tw = 0.0f;
    if (any_ewa) {
#pragma unroll
        for (int t = 0; t < NTAP; t++) {
            const int j = t / WIN, k = t % WIN;
            const int base = rows[j] * W_IN + cols[k];
            const float q1 = relx[k] * cmaj_x + rely[j] * cmaj_y;
            const float q2 = relx[k] * cmin_x + rely[j] * cmin_y;
            const float we = robidoux(q1 * q1 + q2 * q2);
            tw += we;
            ewa[0] += img[base] * we;
            ewa[1] += img[base + H_IN * W_IN] * we;
            ewa[2] += img[base + 2 * H_IN * W_IN] * we;
        }
    }
    const float rtw = __builtin_amdgcn_rcpf(tw + 1e-8f);

    // ---- Combine + store ---------------------------------------------------
#pragma unroll
    for (int ch = 0; ch < NCH; ch++) {
        float mv  = ext_sigmoid(msum[ch], rsig1h, slope);
        float ev  = ewa[ch] * rtw;
        float res = need_ewa ? (theta * mv + (1.0f - theta) * ev) : mv;
        out[(ch * H_OUT + y) * W_OUT + x] = res;
    }
}

extern "C" void kernel_launch(void* const* d_in, const int* in_sizes, int n_in,
                              void* d_out, int out_size, void* d_ws, size_t ws_size,
                              hipStream_t stream) {
    const float* img  = (const float*)d_in[0];   // (1,3,1024,1024) f32
    const float* grd  = (const float*)d_in[1];   // (1,768,768,2)   f32
    float*       outp = (float*)d_out;           // (1,3,768,768)   f32
    (void)in_sizes; (void)n_in; (void)out_size; (void)d_ws; (void)ws_size;

    dim3 gridDims(W_OUT / 256, H_OUT, 1);        // 3 x 768 blocks, full EXEC
    LohaloKernelScalingSampler_57208964383227_kernel<<<gridDims, 256, 0, stream>>>(img, grd, outp);
}